// TimeConv_72086731096515
// MI455X (gfx1250) — compile-verified
//
#include <hip/hip_runtime.h>

#define L 8
#define M 8192
#define D 16
#define H 256
#define F 64
#define N_TOT (L * M)
#define KMAX 512

typedef float v2f __attribute__((ext_vector_type(2)));
typedef float v8f __attribute__((ext_vector_type(8)));

__device__ __forceinline__ float leaky(float x) { return x > 0.f ? x : 0.1f * x; }

// ---------------------------------------------------------------------------
// f32 WMMA GEMM, LDS-staged W panel:
//   C[r, ccol0+c] = act( sum_k A[r,k]*W[k,c] + b[c] )
// Block = 128 threads (4 waves). Each block computes a 64x16 output tile:
// the K x 16 panel of W is staged ONCE into LDS (transposed, stride KMAX+2 so
// every v2f read is 8B-aligned), then each wave runs V_WMMA_F32_16X16X4_F32
// over its 16-row slice. A is read as coalesced global b64 per lane.
// A-matrix lane layout (ISA 7.12.2): lanes 0-15 hold K=0,1; lanes 16-31 K=2,3.
// act: 0 = none, 1 = leaky(0.1). po != null: relu rows where po[r]==0.
// ---------------------------------------------------------------------------
__global__ __launch_bounds__(128)
void gemm_wmma(const float* __restrict__ A, const float* __restrict__ W,
               const float* __restrict__ bias, float* __restrict__ C,
               int K, int Nw, int ldc, int ccol0, int act,
               const int* __restrict__ po)
{
    __shared__ float sW[16][KMAX + 2];   // transposed W panel: sW[col][k]

    const int t    = threadIdx.x;
    const int lane = t & 31;
    const int wave = t >> 5;
    const int half = lane >> 4;          // 0: lanes 0-15, 1: lanes 16-31
    const int lm   = lane & 15;
    const int col0 = blockIdx.x * 16;
    const int row0 = blockIdx.y * 64 + wave * 16;

    // Cooperative stage of W[K x 16] panel (coalesced 16-float row segments).
    for (int i = t; i < K * 16; i += 128) {
        int k = i >> 4, c = i & 15;
        sW[c][k] = W[(size_t)k * Nw + col0 + c];
    }
    __syncthreads();

    v8f acc = {0.f, 0.f, 0.f, 0.f, 0.f, 0.f, 0.f, 0.f};
    const float* arow = A + (size_t)(row0 + lm) * K;
    for (int k = 0; k < K; k += 4) {
        v2f a  = *(const v2f*)(arow + k + 2 * half);       // global b64
        v2f bm = *(const v2f*)(&sW[lm][k + 2 * half]);     // ds_load_b64
        acc = __builtin_amdgcn_wmma_f32_16x16x4_f32(
            /*neg_a=*/false, a, /*neg_b=*/false, bm,
            /*c_mod=*/(short)0, acc, /*reuse_a=*/false, /*reuse_b=*/false);
    }
    const int   c  = col0 + lm;
    const float bb = bias[c];
#pragma unroll
    for (int g = 0; g < 8; ++g) {
        int   r = row0 + g + half * 8;   // D layout: VGPR g -> row g (+8 upper lanes)
        float v = acc[g] + bb;
        if (act) v = leaky(v);
        if (po && po[r] == 0) v = fmaxf(v, 0.f);
        C[(size_t)r * ldc + ccol0 + c] = v;
    }
}

// ---------------------------------------------------------------------------
// Per-level fused attention: gather 16 neighbor rows (H=256) straight into
// LDS via GLOBAL_LOAD_ASYNC_TO_LDS_B32 (no VGPR round-trip, ASYNCcnt-tracked),
// e[d] = h_src[d]·av_h + t[m] + p[m,d]; softmax over d; weighted sum -> neigh.
// One 256-thread block per destination node.
// ---------------------------------------------------------------------------
__global__ __launch_bounds__(256)
void attn_kernel(const float* __restrict__ h_all,
                 const int*   __restrict__ nbr,    // M*D for this level
                 const float* __restrict__ tsc,    // M  for this level
                 const float* __restrict__ pterm,  // M*D for this level
                 const float* __restrict__ av_h,   // 256
                 float*       __restrict__ neigh)  // M x 256
{
    __shared__ float hsmem[D][H];   // 16 KB tile
    __shared__ int   idx_sh[D];
    __shared__ float e_sh[D];
    __shared__ float alpha_sh[D];

    const int m = blockIdx.x;
    const int t = threadIdx.x;

    if (t < D) idx_sh[t] = nbr[m * D + t];
    __syncthreads();

    // Async gather: per-lane LDS byte offset (generic-pointer low 32 bits are
    // the LDS offset per the flat-aperture rule) <- global element address.
#pragma unroll
    for (int d = 0; d < D; ++d) {
        unsigned int       ldsoff = (unsigned int)(uintptr_t)(&hsmem[d][t]);
        unsigned long long gaddr  =
            (unsigned long long)(const void*)(h_all + (size_t)idx_sh[d] * H + t);
        asm volatile("global_load_async_to_lds_b32 %0, %1, off"
                     :: "v"(ldsoff), "v"(gaddr) : "memory");
    }
    asm volatile("s_wait_asynccnt 0x0" ::: "memory");
    __syncthreads();

    // 16 lanes per neighbor: partial dot over a 16-wide chunk, reduce in half-wave
    {
        int   d  = t >> 4;
        int   j0 = (t & 15) * 16;
        float p  = 0.f;
#pragma unroll
        for (int j = 0; j < 16; ++j)
            p += hsmem[d][j0 + j] * av_h[j0 + j];
        p += __shfl_xor(p, 8, 16);
        p += __shfl_xor(p, 4, 16);
        p += __shfl_xor(p, 2, 16);
        p += __shfl_xor(p, 1, 16);
        if ((t & 15) == 0) e_sh[d] = p;
    }
    __syncthreads();

    // softmax over D=16 (lanes 0-15 of wave 0)
    if (t < D) {
        float e  = e_sh[t] + tsc[m] + pterm[m * D + t];
        float mx = e;
        mx = fmaxf(mx, __shfl_xor(mx, 8, 16));
        mx = fmaxf(mx, __shfl_xor(mx, 4, 16));
        mx = fmaxf(mx, __shfl_xor(mx, 2, 16));
        mx = fmaxf(mx, __shfl_xor(mx, 1, 16));
        float ex = __expf(e - mx);
        float s  = ex;
        s += __shfl_xor(s, 8, 16);
        s += __shfl_xor(s, 4, 16);
        s += __shfl_xor(s, 2, 16);
        s += __shfl_xor(s, 1, 16);
        alpha_sh[t] = ex / s;
    }
    __syncthreads();

    float acc = 0.f;
#pragma unroll
    for (int d = 0; d < D; ++d)
        acc += alpha_sh[d] * hsmem[d][t];
    neigh[(size_t)m * H + t] = acc;
}

// ---------------------------------------------------------------------------
// t-term precompute: for rows M..N-1: (feat row -> 32 leaky -> 32) · av_t
// ---------------------------------------------------------------------------
__global__ __launch_bounds__(256)
void tscalar_kernel(const float* __restrict__ feat,
                    const float* __restrict__ W1, const float* __restrict__ b1,
                    const float* __restrict__ W2, const float* __restrict__ b2,
                    const float* __restrict__ av_t,
                    float* __restrict__ out)
{
    __shared__ float sW1[F * 32];
    __shared__ float sW2[32 * 32];
    __shared__ float sb1[32], sb2[32], sav[32];
    const int t = threadIdx.x;
    for (int i = t; i < F * 32; i += 256) sW1[i] = W1[i];
    for (int i = t; i < 32 * 32; i += 256) sW2[i] = W2[i];
    if (t < 32) { sb1[t] = b1[t]; sb2[t] = b2[t]; sav[t] = av_t[t]; }
    __syncthreads();

    const int    row = blockIdx.x * 256 + t;          // 0 .. N_TOT-M-1
    const float* f   = feat + (size_t)(M + row) * F;

    float h1[32];
#pragma unroll
    for (int o = 0; o < 32; ++o) h1[o] = sb1[o];
    for (int j = 0; j < F; ++j) {
        float fj = f[j];
#pragma unroll
        for (int o = 0; o < 32; ++o) h1[o] += fj * sW1[j * 32 + o];
    }
#pragma unroll
    for (int o = 0; o < 32; ++o) h1[o] = leaky(h1[o]);

    float p = 0.f;
#pragma unroll
    for (int o = 0; o < 32; ++o) {
        float a = sb2[o];
#pragma unroll
        for (int j = 0; j < 32; ++j) a += h1[j] * sW2[j * 32 + o];
        p += a * sav[o];
    }
    out[row] = p;
}

// ---------------------------------------------------------------------------
// p-term precompute: per (level,node,neighbor) scalar x: (x ->32 leaky ->32)·av_p
// ---------------------------------------------------------------------------
__global__ __launch_bounds__(256)
void pterm_kernel(const float* __restrict__ bit_pos,
                  const float* __restrict__ W1, const float* __restrict__ b1,
                  const float* __restrict__ W2, const float* __restrict__ b2,
                  const float* __restrict__ av_p,
                  float* __restrict__ out)
{
    __shared__ float sW2[32 * 32];
    __shared__ float sW1[32], sb1[32], sb2[32], sav[32];
    const int t = threadIdx.x;
    for (int i = t; i < 32 * 32; i += 256) sW2[i] = W2[i];
    if (t < 32) { sW1[t] = W1[t]; sb1[t] = b1[t]; sb2[t] = b2[t]; sav[t] = av_p[t]; }
    __syncthreads();

    const size_t id = (size_t)blockIdx.x * 256 + t;
    const float  x  = bit_pos[id];

    float h1[32];
#pragma unroll
    for (int o = 0; o < 32; ++o) h1[o] = leaky(x * sW1[o] + sb1[o]);

    float p = 0.f;
#pragma unroll
    for (int o = 0; o < 32; ++o) {
        float a = sb2[o];
#pragma unroll
        for (int j = 0; j < 32; ++j) a += h1[j] * sW2[j * 32 + o];
        p += a * sav[o];
    }
    out[id] = p;
}

// Elementwise first layer of a scalar-input MLP: out[m,j] = leaky(x[m]*W1[j]+b1[j])
__global__ void scalar_layer(const float* __restrict__ x, const float* __restrict__ W1,
                             const float* __restrict__ b1, float* __restrict__ out, int Hh)
{
    int id = blockIdx.x * blockDim.x + threadIdx.x;  // over M*Hh
    int m = id / Hh, j = id % Hh;
    out[id] = leaky(x[m] * W1[j] + b1[j]);
}

// Copy h7 rows (M x 256) into concat buffer (M x 512), columns [0,256)
__global__ void copy_rows(const float* __restrict__ src, float* __restrict__ dst)
{
    int id = blockIdx.x * blockDim.x + threadIdx.x;  // over M*H
    int m = id / H, j = id % H;
    dst[(size_t)m * 512 + j] = src[id];
}

// Final: out[m] = b_o2 + hid[m,:]·W_o2
__global__ __launch_bounds__(256)
void final_out(const float* __restrict__ hid, const float* __restrict__ W_o2,
               const float* __restrict__ b_o2, float* __restrict__ out)
{
    __shared__ float red[256];
    const int t = threadIdx.x;
    const int m = blockIdx.x;
    red[t] = hid[(size_t)m * H + t] * W_o2[t];
    __syncthreads();
    for (int s = 128; s > 0; s >>= 1) {
        if (t < s) red[t] += red[t + s];
        __syncthreads();
    }
    if (t == 0) out[m] = red[0] + b_o2[0];
}

extern "C" void kernel_launch(void* const* d_in, const int* in_sizes, int n_in,
                              void* d_out, int out_size, void* d_ws, size_t ws_size,
                              hipStream_t stream)
{
    (void)in_sizes; (void)n_in; (void)out_size; (void)ws_size;

    const float* W_pi1 = (const float*)d_in[0];
    const float* b_pi1 = (const float*)d_in[1];
    const float* W_pi2 = (const float*)d_in[2];
    const float* b_pi2 = (const float*)d_in[3];
    const float* W_t1  = (const float*)d_in[4];
    const float* b_t1  = (const float*)d_in[5];
    const float* W_t2  = (const float*)d_in[6];
    const float* b_t2  = (const float*)d_in[7];
    const float* W_p1  = (const float*)d_in[8];
    const float* b_p1  = (const float*)d_in[9];
    const float* W_p2  = (const float*)d_in[10];
    const float* b_p2  = (const float*)d_in[11];
    const float* W_n1  = (const float*)d_in[12];
    const float* b_n1  = (const float*)d_in[13];
    const float* W_n2  = (const float*)d_in[14];
    const float* b_n2  = (const float*)d_in[15];
    const float* W_g1  = (const float*)d_in[16];
    const float* b_g1  = (const float*)d_in[17];
    const float* W_g2  = (const float*)d_in[18];
    const float* b_g2  = (const float*)d_in[19];
    const float* W_o1  = (const float*)d_in[20];
    const float* b_o1  = (const float*)d_in[21];
    const float* W_o2  = (const float*)d_in[22];
    const float* b_o2  = (const float*)d_in[23];
    const float* av    = (const float*)d_in[24];
    const float* feat  = (const float*)d_in[25];
    const float* delay = (const float*)d_in[26];
    const float* bitp  = (const float*)d_in[27];
    const float* po_f  = (const float*)d_in[28];
    const int*   is_po = (const int*)d_in[29];
    const int*   nbr   = (const int*)d_in[30];

    const float* av_t = av;
    const float* av_p = av + 32;
    const float* av_h = av + 64;

    // Workspace layout (floats): ~96 MB total; h_all stays L2-resident (192 MB L2)
    float* ws    = (float*)d_ws;
    float* h_all = ws;                                  // N_TOT*H
    float* tsc   = h_all + (size_t)N_TOT * H;           // N_TOT-M
    float* ptm   = tsc + (N_TOT - M);                   // (L-1)*M*D
    float* bufA  = ptm + (size_t)(L - 1) * M * D;       // M*512 (neigh / concat)
    float* bufB  = bufA + (size_t)M * 512;              // M*256 (hidden)

    // Level-invariant attention terms
    tscalar_kernel<<<(N_TOT - M) / 256, 256, 0, stream>>>(feat, W_t1, b_t1, W_t2, b_t2, av_t, tsc);
    pterm_kernel<<<((L - 1) * M * D) / 256, 256, 0, stream>>>(bitp, W_p1, b_p1, W_p2, b_p2, av_p, ptm);

    // Level 0: h_all[0:M] = MLP_pi(delay)   (scalar layer + WMMA GEMM 128->256)
    scalar_layer<<<(M * 128) / 256, 256, 0, stream>>>(delay, W_pi1, b_pi1, bufB, 128);
    gemm_wmma<<<dim3(H / 16, M / 64), 128, 0, stream>>>(bufB, W_pi2, b_pi2, h_all,
                                                        128, H, H, 0, /*act=*/0, nullptr);

    // Levels 1..7 (sequential dependency)
    for (int i = 1; i < L; ++i) {
        attn_kernel<<<M, 256, 0, stream>>>(h_all, nbr + (size_t)(i - 1) * M * D,
                                           tsc + (size_t)(i - 1) * M,
                                           ptm + (size_t)(i - 1) * M * D, av_h, bufA);
        gemm_wmma<<<dim3(128 / 16, M / 64), 128, 0, stream>>>(bufA, W_n1, b_n1, bufB,
                                                              H, 128, 128, 0, /*act=*/1, nullptr);
        gemm_wmma<<<dim3(H / 16, M / 64), 128, 0, stream>>>(bufB, W_n2, b_n2,
                                                            h_all + (size_t)i * M * H,
                                                            128, H, H, 0, /*act=*/0, is_po + i * M);
    }

    // h_global = MLP_g(po_feat); write directly into concat columns [256,512)
    scalar_layer<<<(M * 128) / 256, 256, 0, stream>>>(po_f, W_g1, b_g1, bufB, 128);
    copy_rows<<<(M * H) / 256, 256, 0, stream>>>(h_all + (size_t)(N_TOT - M) * H, bufA);
    gemm_wmma<<<dim3(H / 16, M / 64), 128, 0, stream>>>(bufB, W_g2, b_g2, bufA,
                                                        128, H, 512, 256, /*act=*/0, nullptr);

    // Output MLP: 512 -> 256 (leaky, WMMA) -> 1 (reduction)
    gemm_wmma<<<dim3(H / 16, M / 64), 128, 0, stream>>>(bufA, W_o1, b_o1, bufB,
                                                        512, H, H, 0, /*act=*/1, nullptr);
    final_out<<<M, 256, 0, stream>>>(bufB, W_o2, b_o2, (float*)d_out);
}